// VQVAEQuantizer_54271206752864
// MI455X (gfx1250) — compile-verified
//
#include <hip/hip_runtime.h>
#include <hip/hip_bf16.h>

#define NUM_EMB 8192
#define EMB_DIM 256
#define FAC     16
#define NPIX    8192      // B*H*W = 8*32*32
#define HW      1024
#define EPSF    1e-6f

typedef float v2f __attribute__((ext_vector_type(2)));
typedef float v8f __attribute__((ext_vector_type(8)));

// ------------------------------------------------------------------
// K1: normalize codebook rows: emb_n[n][d] = emb_w[n][d] / max(||row||, eps)
// ------------------------------------------------------------------
__global__ void k_norm_codebook(const float* __restrict__ emb_w,
                                float* __restrict__ emb_n) {
    int row = blockIdx.x * blockDim.x + threadIdx.x;
    const float4* src = (const float4*)(emb_w + (size_t)row * FAC);
    float4 a = src[0], b = src[1], c = src[2], d = src[3];
    float s = a.x*a.x + a.y*a.y + a.z*a.z + a.w*a.w
            + b.x*b.x + b.y*b.y + b.z*b.z + b.w*b.w
            + c.x*c.x + c.y*c.y + c.z*c.z + c.w*c.w
            + d.x*d.x + d.y*d.y + d.z*d.z + d.w*d.w;
    float inv = 1.0f / fmaxf(sqrtf(s), EPSF);
    float4* dst = (float4*)(emb_n + (size_t)row * FAC);
    a.x*=inv; a.y*=inv; a.z*=inv; a.w*=inv;
    b.x*=inv; b.y*=inv; b.z*=inv; b.w*=inv;
    c.x*=inv; c.y*=inv; c.z*=inv; c.w*=inv;
    d.x*=inv; d.y*=inv; d.z*=inv; d.w*=inv;
    dst[0]=a; dst[1]=b; dst[2]=c; dst[3]=d;
}

// ------------------------------------------------------------------
// K2: projection (1x1 conv) + L2-normalize per pixel.
// thread = pixel; channel reads coalesced (stride HW across channels);
// proj_w transposed into LDS so inner reads are broadcast b128 loads.
// ------------------------------------------------------------------
__global__ void k_project(const float* __restrict__ encoded,
                          const float* __restrict__ proj_w,   // [16][256]
                          const float* __restrict__ proj_b,   // [16]
                          float* __restrict__ pre,            // [NPIX][16]
                          float* __restrict__ enc_n) {        // [NPIX][16]
    __shared__ __align__(16) float lds_pwT[EMB_DIM * FAC];    // [c][d]
    int t = threadIdx.x;
    #pragma unroll
    for (int d = 0; d < FAC; ++d)
        lds_pwT[t * FAC + d] = proj_w[d * EMB_DIM + t];       // coalesced per d
    __syncthreads();

    int p  = blockIdx.x * blockDim.x + t;
    int bb = p >> 10, hw = p & (HW - 1);
    const float* g = encoded + (size_t)bb * EMB_DIM * HW + hw;

    float acc[FAC];
    #pragma unroll
    for (int d = 0; d < FAC; ++d) acc[d] = proj_b[d];

    for (int c = 0; c < EMB_DIM; ++c) {
        float x = g[(size_t)c << 10];
        const float* wr = &lds_pwT[c * FAC];
        float4 w0 = *(const float4*)(wr + 0);
        float4 w1 = *(const float4*)(wr + 4);
        float4 w2 = *(const float4*)(wr + 8);
        float4 w3 = *(const float4*)(wr + 12);
        acc[ 0]=fmaf(x,w0.x,acc[ 0]); acc[ 1]=fmaf(x,w0.y,acc[ 1]);
        acc[ 2]=fmaf(x,w0.z,acc[ 2]); acc[ 3]=fmaf(x,w0.w,acc[ 3]);
        acc[ 4]=fmaf(x,w1.x,acc[ 4]); acc[ 5]=fmaf(x,w1.y,acc[ 5]);
        acc[ 6]=fmaf(x,w1.z,acc[ 6]); acc[ 7]=fmaf(x,w1.w,acc[ 7]);
        acc[ 8]=fmaf(x,w2.x,acc[ 8]); acc[ 9]=fmaf(x,w2.y,acc[ 9]);
        acc[10]=fmaf(x,w2.z,acc[10]); acc[11]=fmaf(x,w2.w,acc[11]);
        acc[12]=fmaf(x,w3.x,acc[12]); acc[13]=fmaf(x,w3.y,acc[13]);
        acc[14]=fmaf(x,w3.z,acc[14]); acc[15]=fmaf(x,w3.w,acc[15]);
    }
    float s = 0.f;
    #pragma unroll
    for (int d = 0; d < FAC; ++d) s = fmaf(acc[d], acc[d], s);
    float inv = 1.0f / fmaxf(sqrtf(s), EPSF);

    float* pr = pre   + (size_t)p * FAC;
    float* en = enc_n + (size_t)p * FAC;
    #pragma unroll
    for (int d = 0; d < FAC; ++d) { pr[d] = acc[d]; en[d] = acc[d] * inv; }
}

// ------------------------------------------------------------------
// K3: fused cosine-sim GEMM + argmax via V_WMMA_F32_16X16X4_F32.
// wave = one 16-pixel tile (M), loops all 8192 codes (N) in 16-code tiles,
// K=16 done as 4 chained wmma (K=4 each). Workgroup (8 waves) stages 128
// codebook rows per chunk in LDS. Argmax kept per-lane in D layout, then
// 16-lane shuffle reduction with jnp.argmax first-occurrence tie-break.
// ------------------------------------------------------------------
#define CHUNK 128
#define RS    20   // LDS row stride (floats): keeps b128 staging aligned, spreads banks

__global__ void k_cos_argmax(const float* __restrict__ enc_n,
                             const float* __restrict__ emb_n,
                             int* __restrict__ idx_out) {
    __shared__ __align__(16) float ldsB[CHUNK * RS];
    int t    = threadIdx.x;
    int lane = t & 31;
    int wave = t >> 5;
    int ptile = blockIdx.x * 8 + wave;
    int m    = lane & 15;
    int half = lane >> 4;

    // A matrix (16 pixels x K=16), ISA 16x4 f32 layout per K-chunk:
    // lane m holds K = kk*4 + half*2 + {0,1}
    v2f a[4];
    #pragma unroll
    for (int kk = 0; kk < 4; ++kk) {
        int kb = kk * 4 + half * 2;
        a[kk] = *(const v2f*)(enc_n + ((size_t)ptile * 16 + m) * FAC + kb);
    }

    float bestv[8];
    int   besti[8];
    #pragma unroll
    for (int r = 0; r < 8; ++r) { bestv[r] = -3.402823466e38f; besti[r] = 0; }

    int srow = t >> 1, shalf = t & 1;   // staging: 2 threads per codebook row
    for (int ch = 0; ch < NUM_EMB / CHUNK; ++ch) {
        __syncthreads();
        {   // stage 128 rows x 16 floats (8 KB) cooperatively
            const float4* src =
                (const float4*)(emb_n + ((size_t)ch * CHUNK + srow) * FAC + shalf * 8);
            float4 x0 = src[0], x1 = src[1];
            float4* dst = (float4*)&ldsB[srow * RS + shalf * 8];
            dst[0] = x0; dst[1] = x1;
        }
        __syncthreads();

        #pragma unroll
        for (int jj = 0; jj < CHUNK / 16; ++jj) {
            v8f acc = {};
            #pragma unroll
            for (int kk = 0; kk < 4; ++kk) {
                int kb = kk * 4 + half * 2;
                v2f b = *(const v2f*)&ldsB[(jj * 16 + m) * RS + kb];
                acc = __builtin_amdgcn_wmma_f32_16x16x4_f32(
                        false, a[kk], false, b, (short)0, acc, false, false);
            }
            // D layout: VGPR r -> pixel row (half? r+8 : r), col = lane&15
            int code = ch * CHUNK + jj * 16 + m;
            #pragma unroll
            for (int r = 0; r < 8; ++r) {
                if (acc[r] > bestv[r]) { bestv[r] = acc[r]; besti[r] = code; }
            }
        }
    }

    // reduce across the 16 lanes holding the same pixel rows (halves disjoint)
    #pragma unroll
    for (int off = 8; off >= 1; off >>= 1) {
        #pragma unroll
        for (int r = 0; r < 8; ++r) {
            float ov = __shfl_xor(bestv[r], off, 32);
            int   oi = __shfl_xor(besti[r], off, 32);
            if (ov > bestv[r] || (ov == bestv[r] && oi < besti[r])) {
                bestv[r] = ov; besti[r] = oi;
            }
        }
    }
    if ((lane & 15) == 0) {
        #pragma unroll
        for (int r = 0; r < 8; ++r)
            idx_out[ptile * 16 + half * 8 + r] = besti[r];
    }
}

// ------------------------------------------------------------------
// K4: gather latents, expansion GEMM (writes 8MB out, coalesced),
// MSE partials (deterministic per-block), usage histogram.
// ------------------------------------------------------------------
__global__ void k_expand(const float* __restrict__ emb_n,
                         const float* __restrict__ pre,
                         const int*   __restrict__ idx,
                         const float* __restrict__ exp_w,   // [256][16]
                         const float* __restrict__ exp_b,   // [256]
                         float* __restrict__ out,
                         float* __restrict__ block_loss,    // [gridDim.x]
                         int*   __restrict__ counts) {
    __shared__ __align__(16) float lds_ew[EMB_DIM * FAC];
    __shared__ float lds_eb[EMB_DIM];
    __shared__ float wsum[8];
    int t = threadIdx.x;
    #pragma unroll
    for (int i = 0; i < FAC; ++i)
        lds_ew[t * FAC + i] = exp_w[t * FAC + i];           // fully coalesced
    lds_eb[t] = exp_b[t];
    __syncthreads();

    int p = blockIdx.x * blockDim.x + t;
    int code = idx[p];
    atomicAdd(&counts[code], 1);

    float lat[FAC];
    const float* er = emb_n + (size_t)code * FAC;
    #pragma unroll
    for (int d = 0; d < FAC; ++d) lat[d] = er[d];

    // loss partial: sum (pre - latent)^2
    const float* pr = pre + (size_t)p * FAC;
    float sq = 0.f;
    #pragma unroll
    for (int d = 0; d < FAC; ++d) { float df = pr[d] - lat[d]; sq = fmaf(df, df, sq); }
    #pragma unroll
    for (int off = 16; off >= 1; off >>= 1) sq += __shfl_xor(sq, off, 32);
    if ((t & 31) == 0) wsum[t >> 5] = sq;
    __syncthreads();
    if (t == 0) {
        float s = 0.f;
        #pragma unroll
        for (int w = 0; w < 8; ++w) s += wsum[w];
        block_loss[blockIdx.x] = s;
    }

    int bb = p >> 10, hw = p & (HW - 1);
    float* o = out + (size_t)bb * EMB_DIM * HW + hw;
    for (int c = 0; c < EMB_DIM; ++c) {
        const float* wr = &lds_ew[c * FAC];
        float4 w0 = *(const float4*)(wr + 0);
        float4 w1 = *(const float4*)(wr + 4);
        float4 w2 = *(const float4*)(wr + 8);
        float4 w3 = *(const float4*)(wr + 12);
        float v = lds_eb[c];
        v=fmaf(lat[ 0],w0.x,v); v=fmaf(lat[ 1],w0.y,v);
        v=fmaf(lat[ 2],w0.z,v); v=fmaf(lat[ 3],w0.w,v);
        v=fmaf(lat[ 4],w1.x,v); v=fmaf(lat[ 5],w1.y,v);
        v=fmaf(lat[ 6],w1.z,v); v=fmaf(lat[ 7],w1.w,v);
        v=fmaf(lat[ 8],w2.x,v); v=fmaf(lat[ 9],w2.y,v);
        v=fmaf(lat[10],w2.z,v); v=fmaf(lat[11],w2.w,v);
        v=fmaf(lat[12],w3.x,v); v=fmaf(lat[13],w3.y,v);
        v=fmaf(lat[14],w3.z,v); v=fmaf(lat[15],w3.w,v);
        o[(size_t)c << 10] = v;
    }
}

// ------------------------------------------------------------------
// K5: perplexity from histogram + final loss scalar. One workgroup.
// ------------------------------------------------------------------
__global__ void k_finalize(const int* __restrict__ counts,
                           const float* __restrict__ block_loss,
                           int n_blocks,
                           float* __restrict__ loss_out,
                           float* __restrict__ perp_out) {
    __shared__ float red[256];
    int t = threadIdx.x;
    float s = 0.f;
    const int per = NUM_EMB / 256;
    for (int i = 0; i < per; ++i) {
        float u = (float)counts[t * per + i] * (1.0f / (float)NPIX);
        s -= u * __logf(u + 1e-6f);
    }
    red[t] = s;
    __syncthreads();
    for (int stride = 128; stride >= 1; stride >>= 1) {
        if (t < stride) red[t] += red[t + stride];
        __syncthreads();
    }
    if (t == 0) {
        *perp_out = __expf(red[0]);
        float ls = 0.f;
        for (int i = 0; i < n_blocks; ++i) ls += block_loss[i];
        *loss_out = ls / (float)(NPIX * FAC);
    }
}

// ------------------------------------------------------------------
extern "C" void kernel_launch(void* const* d_in, const int* in_sizes, int n_in,
                              void* d_out, int out_size, void* d_ws, size_t ws_size,
                              hipStream_t stream) {
    const float* encoded = (const float*)d_in[0];
    const float* emb_w   = (const float*)d_in[1];
    const float* proj_w  = (const float*)d_in[2];
    const float* proj_b  = (const float*)d_in[3];
    const float* exp_w   = (const float*)d_in[4];
    const float* exp_b   = (const float*)d_in[5];

    char* ws = (char*)d_ws;
    float* emb_n      = (float*)(ws);                               // 512 KB
    float* pre        = (float*)(ws + (size_t)512 * 1024);          // 512 KB
    float* enc_n      = (float*)(ws + (size_t)1024 * 1024);         // 512 KB
    int*   counts     = (int*)  (ws + (size_t)1536 * 1024);         // 32 KB
    float* block_loss = (float*)(ws + (size_t)1536 * 1024 + 32 * 1024); // 32 floats

    float* out      = (float*)d_out;                                 // [8,256,32,32]
    int*   idx_out  = (int*)((float*)d_out + (size_t)NPIX * EMB_DIM);// [8192] int32
    float* loss_out = (float*)d_out + (size_t)NPIX * EMB_DIM + NPIX;
    float* perp_out = loss_out + 1;

    hipMemsetAsync(counts, 0, 32 * 1024, stream);

    k_norm_codebook<<<NUM_EMB / 256, 256, 0, stream>>>(emb_w, emb_n);
    k_project     <<<NPIX / 256,     256, 0, stream>>>(encoded, proj_w, proj_b, pre, enc_n);
    k_cos_argmax  <<<NPIX / 16 / 8,  256, 0, stream>>>(enc_n, emb_n, idx_out);
    k_expand      <<<NPIX / 256,     256, 0, stream>>>(emb_n, pre, idx_out, exp_w, exp_b,
                                                       out, block_loss, counts);
    k_finalize    <<<1,              256, 0, stream>>>(counts, block_loss, NPIX / 256,
                                                       loss_out, perp_out);
}